// NoIntraSetLayer_58394375357150
// MI455X (gfx1250) — compile-verified
//
#include <hip/hip_runtime.h>

// ---------------------------------------------------------------------------
// KNN (k=16) over 8 clouds x 8192 pts, then gather. MI455X / gfx1250.
//
// d2(m,n) = |pm|^2 + |pn|^2 - 2 pm.pn
//         = [xm,ym,zm,sqm] . [-2xn,-2yn,-2zn,1]  +  sqn
// -> V_WMMA_F32_16X16X4_F32 with A = candidate tile (M), B = query tile (N),
//    C preloaded with the column-constant sqn (each lane owns query N=lane%16,
//    so C is that lane's query sq replicated in all 8 accumulator VGPRs).
// Each lane receives 8 candidate distances per WMMA for *its* query and keeps
// a sorted top-16 in registers (guarded branchless bubble insert). Lane pairs
// (l, l+16) see disjoint candidate halves; merged via LDS at the end.
//
// Round-2 tuning (from asm inspection):
//  * software-pipelined ds_load_b64 of the next A tile so s_wait_dscnt is
//    covered by a full iteration of independent work, and
//  * v_min tree + single wave-level early-out instead of 8 cmpx/branch pairs
//    on the (common) no-insert path.
// ---------------------------------------------------------------------------

typedef __attribute__((ext_vector_type(2))) float v2f;
typedef __attribute__((ext_vector_type(8))) float v8f;

#define BATCH 8
#define NPTS  8192
#define FEAT  64
#define NS    16
#define CHUNK 2048                        // points staged in LDS (32KB float4)
#define WAVES_PER_BLOCK 8                 // 256 threads = 8 wave32
#define TILES_PER_CLOUD (NPTS / 16)       // 512 query tiles per cloud
#define BLOCKS_PER_CLOUD (TILES_PER_CLOUD / WAVES_PER_BLOCK)  // 64

__global__ __launch_bounds__(256) void knn_wmma_kernel(
    const float* __restrict__ p, float* __restrict__ knn_idx_out) {
  __shared__ float4 s_pts[CHUNK];  // 32KB staging; reused as merge scratch

  const int tid  = threadIdx.x;
  const int lane = tid & 31;
  const int wave = tid >> 5;
  const int cloud = blockIdx.x / BLOCKS_PER_CLOUD;
  const int tile  = (blockIdx.x % BLOCKS_PER_CLOUD) * WAVES_PER_BLOCK + wave;
  const int q_local  = tile * 16 + (lane & 15);
  const int q_global = cloud * NPTS + q_local;

  // ---- query registers: B matrix (4x16, queries along N) + C (= sq_n) ----
  const float qx = p[q_global * 3 + 0];
  const float qy = p[q_global * 3 + 1];
  const float qz = p[q_global * 3 + 2];
  const float qsq = qx * qx + qy * qy + qz * qz;
  v2f bmat;  // VGPR0: K0 (lanes0-15) / K2 (lanes16-31); VGPR1: K1 / K3
  if (lane < 16) { bmat.x = -2.0f * qx; bmat.y = -2.0f * qy; }
  else           { bmat.x = -2.0f * qz; bmat.y = 1.0f;       }
  v8f cacc;
#pragma unroll
  for (int j = 0; j < 8; ++j) cacc[j] = qsq;  // column-constant sq_n

  // ---- per-lane sorted top-16 (ascending; dl[15] = current worst) ----
  float dl[NS];
  int   il[NS];
#pragma unroll
  for (int j = 0; j < NS; ++j) { dl[j] = 3.4e38f; il[j] = 0; }

  // Per-lane A-tile fetch address pattern (conflict-free b64 across 64 banks):
  // lanes0-15 read (x,y) of candidate M=lane, lanes16-31 read (z,sq) of M-16.
  const int asel = (lane & 15) * 2 + (lane >> 4);

#pragma unroll 1
  for (int ch = 0; ch < NPTS / CHUNK; ++ch) {
    // Stage chunk: float4(x,y,z,sq) per point.
#pragma unroll 1
    for (int t = tid; t < CHUNK; t += 256) {
      const int g = (cloud * NPTS + ch * CHUNK + t) * 3;
      const float x = p[g], y = p[g + 1], z = p[g + 2];
      s_pts[t] = make_float4(x, y, z, x * x + y * y + z * z);
      if (ch + 1 < NPTS / CHUNK)  // global_prefetch_b8 next chunk
        __builtin_prefetch(&p[g + 3 * CHUNK], 0, 0);
    }
    __syncthreads();

    const float2* sp2 = (const float2*)s_pts;
    float2 av = sp2[asel];  // pipelined preload of tile ct=0
#pragma unroll 1
    for (int ct = 0; ct < CHUNK / 16; ++ct) {
      v2f amat; amat.x = av.x; amat.y = av.y;
      if (ct + 1 < CHUNK / 16)          // prefetch next A tile; the dscnt
        av = sp2[(ct + 1) * 32 + asel]; // wait is hidden behind this iter

      v8f d = __builtin_amdgcn_wmma_f32_16x16x4_f32(
          false, amat, false, bmat, (short)0, cacc, false, false);

      // Cheap wave-level early-out: min of this lane's 8 candidates.
      const float m01 = fminf(d[0], d[1]);
      const float m23 = fminf(d[2], d[3]);
      const float m45 = fminf(d[4], d[5]);
      const float m67 = fminf(d[6], d[7]);
      const float mall = fminf(fminf(m01, m23), fminf(m45, m67));

      if (mall < dl[NS - 1]) {          // rare once lists converge
        // This lane's 8 candidates: rows M = v + 8*(lane>=16).
        const int cbase = ch * CHUNK + ct * 16 + ((lane >> 4) << 3);
#pragma unroll
        for (int v = 0; v < 8; ++v) {
          const float dv = d[v];
          if (dv < dl[NS - 1]) {
            float cd = dv; int ci = cbase + v;
#pragma unroll
            for (int j = 0; j < NS; ++j) {  // branchless bubble insert
              const bool sm = cd < dl[j];
              const float tv = dl[j]; const int ti = il[j];
              dl[j] = sm ? cd : tv;  il[j] = sm ? ci : ti;
              cd    = sm ? tv : cd;  ci    = sm ? ti : ci;
            }
          }
        }
      }
    }
    __syncthreads();
  }

  // ---- merge half-lists of lane pair (l, l+16) via LDS (reuse staging) ----
  float* md = ((float*)s_pts) + wave * 1024;  // 512 d + 512 idx per wave
  int*   mi = (int*)(md + 512);
#pragma unroll
  for (int j = 0; j < NS; ++j) {
    md[lane * NS + j] = dl[j];
    mi[lane * NS + j] = il[j];
  }
  __syncthreads();

  if (lane < 16) {
    const float* da = md + lane * NS;
    const float* db = md + (lane + 16) * NS;
    const int*   ia = mi + lane * NS;
    const int*   ib = mi + (lane + 16) * NS;
    int i0 = 0, i1 = 0;
    float* outp = knn_idx_out + (size_t)q_global * NS;
#pragma unroll 1
    for (int k = 0; k < NS; ++k) {
      const float va = da[i0], vb = db[i1];
      int gi;
      if (va <= vb) { gi = ia[i0]; ++i0; } else { gi = ib[i1]; ++i1; }
      outp[k] = (float)(cloud * NPTS + gi);  // global index, pointops style
    }
  }
}

// ---- phase 2: pure-bandwidth copy + gathers (outputs 1, 2, 4) -------------

__global__ __launch_bounds__(256) void copy_x_kernel(
    const float4* __restrict__ x4, float4* __restrict__ out4) {
  const size_t i = (size_t)blockIdx.x * 256 + threadIdx.x;
  out4[i] = x4[i];
}

__global__ __launch_bounds__(256) void gather_kernel(
    const float* __restrict__ p, const float4* __restrict__ x4,
    const float* __restrict__ knn_idx, float4* __restrict__ xknn4,
    float* __restrict__ pr) {
  const int tid = threadIdx.x;
  const int r   = blockIdx.x * 16 + (tid >> 4);  // row = q*16 + k
  const int c4  = tid & 15;
  const int gidx = (int)knn_idx[r];
  xknn4[(size_t)r * 16 + c4] = x4[(size_t)gidx * 16 + c4];
  if (c4 == 0) {
    const int q = r >> 4;
    pr[(size_t)r * 3 + 0] = p[gidx * 3 + 0] - p[q * 3 + 0];
    pr[(size_t)r * 3 + 1] = p[gidx * 3 + 1] - p[q * 3 + 1];
    pr[(size_t)r * 3 + 2] = p[gidx * 3 + 2] - p[q * 3 + 2];
  }
}

extern "C" void kernel_launch(void* const* d_in, const int* in_sizes, int n_in,
                              void* d_out, int out_size, void* d_ws,
                              size_t ws_size, hipStream_t stream) {
  const float* p = (const float*)d_in[0];
  const float* x = (const float*)d_in[1];
  // outputs, concatenated flat: x, x_knn, knn_idx (as float), p_r
  float* out      = (float*)d_out;
  float* out_x    = out;
  float* out_xknn = out + (size_t)BATCH * NPTS * FEAT;
  float* out_kidx = out_xknn + (size_t)BATCH * NPTS * NS * FEAT;
  float* out_pr   = out_kidx + (size_t)BATCH * NPTS * NS;

  knn_wmma_kernel<<<BATCH * BLOCKS_PER_CLOUD, 256, 0, stream>>>(p, out_kidx);
  copy_x_kernel<<<(BATCH * NPTS * FEAT / 4) / 256, 256, 0, stream>>>(
      (const float4*)x, (float4*)out_x);
  gather_kernel<<<BATCH * NPTS * NS / 16, 256, 0, stream>>>(
      p, (const float4*)x, out_kidx, (float4*)out_xknn, out_pr);
}